// SmallGNN_25271587570231
// MI455X (gfx1250) — compile-verified
//
#include <hip/hip_runtime.h>
#include <stdint.h>

typedef __attribute__((ext_vector_type(2))) float v2f;
typedef __attribute__((ext_vector_type(8))) float v8f;

#define N_NODES 50000
#define D_IN 128
#define D_HID 256
#define D_OUT 128

// ---------------------------------------------------------------- zero fill
__global__ void gnn_zero_kernel(float* __restrict__ p, long long n) {
  long long i = (long long)blockIdx.x * blockDim.x + threadIdx.x;
  long long stride = (long long)gridDim.x * blockDim.x;
  for (; i < n; i += stride) p[i] = 0.0f;
}

// ---------------------------------------------------------------- degrees
__global__ void gnn_degree_kernel(const long long* __restrict__ dst,
                                  float* __restrict__ cnt, int E) {
  int e = blockIdx.x * blockDim.x + threadIdx.x;
  if (e < E) atomicAdd(&cnt[(int)dst[e]], 1.0f);
}

// ---------------------------------------------------------------- scatter-add
// One lane handles 4 consecutive features of one edge (float4 gather + 4 f32
// atomics into L2-resident agg). LPE >= 32 so every wave works on a single
// edge: readfirstlane scalarizes src/dst into SGPRs (uniform base address).
template <int DIM>
__global__ void gnn_scatter_kernel(const float* __restrict__ feat,
                                   const long long* __restrict__ src,
                                   const long long* __restrict__ dst,
                                   float* __restrict__ agg, int E) {
  const int LPE = DIM / 4;  // lanes per edge (32 or 64)
  long long t = (long long)blockIdx.x * blockDim.x + threadIdx.x;
  int e = (int)(t / LPE);
  int ch = (int)(t % LPE);
  if (e >= E) return;
  int s = __builtin_amdgcn_readfirstlane((int)src[e]);  // wave-uniform
  int d = __builtin_amdgcn_readfirstlane((int)dst[e]);  // wave-uniform
  const float4 v = *(const float4*)(feat + (size_t)s * DIM + ch * 4);
  float* o = agg + (size_t)d * DIM + ch * 4;
  atomicAdd(o + 0, v.x);
  atomicAdd(o + 1, v.y);
  atomicAdd(o + 2, v.z);
  atomicAdd(o + 3, v.w);
}

// ---------------------------------------------------------------- fused SAGE GEMM
// out[m, n] = relu?( (agg[m,:]/max(cnt[m],1)) . Wl[n,:] + self[m,:] . Wr[n,:] + b[n] )
// One wave32 computes a 16x16 output tile via V_WMMA_F32_16X16X4_F32.
// f32 ISA layouts (CDNA5 ISA):
//   A 16x4 : lanes 0-15 -> M=lane, VGPR0=K0, VGPR1=K1 ; lanes 16-31 -> K2,K3
//   B 4x16 : lanes 0-15 -> N=lane, VGPR0=K0, VGPR1=K1 ; lanes 16-31 -> K2,K3
//   C/D    : lane L, VGPR v -> M = v + 8*(L>>4), N = L&15
// Loop control is scalar (k = 0..DIN/4); the lane-dependent K-pair select
// (half) only perturbs the load index, so no divergent-loop exec masking.
template <int DIN, int DOUT, bool RELU>
__global__ __launch_bounds__(128)
void gnn_sage_gemm_kernel(const float* __restrict__ Aagg,
                          const float* __restrict__ Aself,
                          const float* __restrict__ cnt,
                          const float* __restrict__ Wl,
                          const float* __restrict__ Wr,
                          const float* __restrict__ bias,
                          float* __restrict__ out) {
  const int lane = threadIdx.x;          // 0..31, one wave per threadIdx.y
  const int half = lane >> 4;            // K-pair select
  const int lr = lane & 15;              // row/col within tile
  const int m0 = blockIdx.x * 16;
  const int n0 = (blockIdx.y * blockDim.y + threadIdx.y) * 16;
  const int m = m0 + lr;
  const int n = n0 + lr;

  const float scale = 1.0f / fmaxf(cnt[m], 1.0f);  // mean (empty -> /1 of zeros)
  const float bval = bias[n];

  v8f c;
#pragma unroll
  for (int v = 0; v < 8; ++v) c[v] = bval;  // D += bias (broadcast over M)

  const float2* __restrict__ Ag = (const float2*)(Aagg + (size_t)m * DIN);
  const float2* __restrict__ As = (const float2*)(Aself + (size_t)m * DIN);
  const float2* __restrict__ Bl = (const float2*)(Wl + (size_t)n * DIN);
  const float2* __restrict__ Br = (const float2*)(Wr + (size_t)n * DIN);

  // Both SAGE terms fused: 2 WMMAs + 4 float2 loads per K-step of 4.
#pragma unroll 4
  for (int k = 0; k < DIN / 4; ++k) {
    const int idx = 2 * k + half;  // float2 index: column k*4 + 2*half
    float2 a1 = Ag[idx];
    float2 w1 = Bl[idx];
    float2 a2 = As[idx];
    float2 w2 = Br[idx];

    v2f av1; av1.x = a1.x * scale; av1.y = a1.y * scale;
    v2f wv1; wv1.x = w1.x; wv1.y = w1.y;
    c = __builtin_amdgcn_wmma_f32_16x16x4_f32(false, av1, false, wv1, (short)0,
                                              c, false, false);

    v2f av2; av2.x = a2.x; av2.y = a2.y;
    v2f wv2; wv2.x = w2.x; wv2.y = w2.y;
    c = __builtin_amdgcn_wmma_f32_16x16x4_f32(false, av2, false, wv2, (short)0,
                                              c, false, false);
  }

#pragma unroll
  for (int v = 0; v < 8; ++v) {
    float val = c[v];
    if (RELU) val = fmaxf(val, 0.0f);
    out[(size_t)(m0 + v + 8 * half) * DOUT + n] = val;
  }
}

// ---------------------------------------------------------------- launcher
extern "C" void kernel_launch(void* const* d_in, const int* in_sizes, int n_in,
                              void* d_out, int out_size, void* d_ws, size_t ws_size,
                              hipStream_t stream) {
  const float* x = (const float*)d_in[0];
  const long long* edge = (const long long*)d_in[1];  // int64 [2, E]
  const float* W1l = (const float*)d_in[2];
  const float* W1r = (const float*)d_in[3];
  const float* b1 = (const float*)d_in[4];
  const float* W2l = (const float*)d_in[5];
  const float* W2r = (const float*)d_in[6];
  const float* b2 = (const float*)d_in[7];
  float* out = (float*)d_out;

  const int E = in_sizes[1] / 2;
  const long long* src = edge;
  const long long* dstp = edge + E;

  // Workspace layout (floats): agg1 | agg2 | cnt | h
  float* ws = (float*)d_ws;
  float* agg1 = ws;                                  // N*128
  float* agg2 = agg1 + (size_t)N_NODES * D_IN;       // N*256
  float* cnt = agg2 + (size_t)N_NODES * D_HID;       // N
  float* h = cnt + N_NODES;                          // N*256 (fully overwritten)

  // Re-zero accumulation regions every call (graph replay safe).
  const long long nz = (long long)N_NODES * (D_IN + D_HID) + N_NODES;
  gnn_zero_kernel<<<4096, 256, 0, stream>>>(ws, nz);

  // In-degrees (shared by both layers; same graph).
  gnn_degree_kernel<<<(E + 255) / 256, 256, 0, stream>>>(dstp, cnt, E);

  // Layer 1 aggregate: agg1[dst] += x[src]
  {
    long long T = (long long)E * (D_IN / 4);
    gnn_scatter_kernel<D_IN><<<(unsigned)((T + 255) / 256), 256, 0, stream>>>(
        x, src, dstp, agg1, E);
  }

  // h = relu(mean(agg1) @ W1_l^T + x @ W1_r^T + b1)
  gnn_sage_gemm_kernel<D_IN, D_HID, true>
      <<<dim3(N_NODES / 16, D_HID / 64), dim3(32, 4), 0, stream>>>(
          agg1, x, cnt, W1l, W1r, b1, h);

  // Layer 2 aggregate: agg2[dst] += h[src]
  {
    long long T = (long long)E * (D_HID / 4);
    gnn_scatter_kernel<D_HID><<<(unsigned)((T + 255) / 256), 256, 0, stream>>>(
        h, src, dstp, agg2, E);
  }

  // out = mean(agg2) @ W2_l^T + h @ W2_r^T + b2
  gnn_sage_gemm_kernel<D_HID, D_OUT, false>
      <<<dim3(N_NODES / 16, D_OUT / 64), dim3(32, 4), 0, stream>>>(
          agg2, h, cnt, W2l, W2r, b2, out);
}